// GAT_79834852098717
// MI455X (gfx1250) — compile-verified
//
#include <hip/hip_runtime.h>
#include <hip/hip_bf16.h>
#include <stdint.h>

#define N_CH_IN  256
#define HIDC     128
#define OUTC     40
#define NEG_SLOPE 0.2f
#define BN_EPS    1e-5f
#define KC       32          // K-chunk depth staged in LDS
#define ROWP     36          // padded A row stride (floats)

typedef __attribute__((ext_vector_type(2))) float v2f;
typedef __attribute__((ext_vector_type(8))) float v8f;

// CDNA5 async global->LDS copy (16B per lane), tracked by ASYNCcnt.
__device__ __forceinline__ void async_ld_b128(void* lds, const void* gaddr)
{
    const uint32_t ldsOff = (uint32_t)(uintptr_t)lds;   // low 32b = LDS offset
    asm volatile("global_load_async_to_lds_b128 %0, %1, off"
                 :: "v"(ldsOff), "v"(gaddr) : "memory");
}
__device__ __forceinline__ void wait_async0()
{
    asm volatile("s_wait_asynccnt 0x0" ::: "memory");
}

// pad columns to NC2 with NC2 % 32 == 16 -> half-wave LDS offset = 32 banks
template<int NC> struct PadCols { static constexpr int v = (NC % 32 == 16) ? NC : NC + 16; };

// ---------------------------------------------------------------------------
// LDS-tiled WMMA fp32 GEMM: C[M, ldC] = A[M,K] @ B (+ bias).
// B is pre-repacked chunk-major & pair-interleaved:
//   Brp[k/32][(k%32)/2][col (NC2)][k%2]   (pads zero-filled)
// so a chunk stages as one flat contiguous async copy and every B fragment is
// a single conflict-free ds_load_b64. A staged row-major [row][ROWP].
// Requires K % 32 == 0, M % 16 == 0.
// ---------------------------------------------------------------------------
template<int NT>
__global__ __launch_bounds__(256)
void gat_wmma_gemm(const float* __restrict__ A, const float* __restrict__ Brp,
                   const float* __restrict__ bias, float* __restrict__ C,
                   int M, int K, int ldC, int Nstore)
{
    constexpr int NC    = 16 * NT;
    constexpr int NC2   = PadCols<NC>::v;
    constexpr int CHUNK = 16 * NC2 * 2;        // floats per repacked B chunk
    __shared__ float bsh[CHUNK];
    __shared__ float ash[128 * ROWP];

    const int wave = threadIdx.x >> 5;
    const int lane = threadIdx.x & 31;
    const int half = lane >> 4;
    const int lr   = lane & 15;

    const int brow0 = blockIdx.x * 128;
    const int bm    = brow0 + wave * 16;
    const int rr    = wave * 16 + lr;

    v8f acc[NT];
    #pragma unroll
    for (int t = 0; t < NT; ++t) acc[t] = (v8f){};

    for (int kc0 = 0; kc0 < K; kc0 += KC) {
        __syncthreads();
        // ---- stage repacked B chunk: flat contiguous copy ----
        {
            const float* Bg = Brp + (size_t)(kc0 / KC) * CHUNK;
            for (int v = threadIdx.x; v < CHUNK / 4; v += 256)
                async_ld_b128(&bsh[v * 4], Bg + v * 4);
        }
        // ---- stage A chunk [128 rows x KC], row-major ----
        {
            constexpr int NVA = 128 * KC / 4;
            for (int v = threadIdx.x; v < NVA; v += 256) {
                const int r  = v >> 3;
                const int kb = (v & 7) * 4;
                int gr = brow0 + r;
                if (gr >= M) gr = M - 1;
                async_ld_b128(&ash[r * ROWP + kb],
                              A + (size_t)gr * K + kc0 + kb);
            }
        }
        wait_async0();
        __syncthreads();

        #pragma unroll
        for (int i = 0; i < KC / 4; ++i) {
            const int p = 2 * i + half;                        // k-pair index
            const v2f a = *(const v2f*)&ash[rr * ROWP + 4 * i + 2 * half];
            v2f bb[NT];
            #pragma unroll
            for (int t = 0; t < NT; ++t)
                bb[t] = *(const v2f*)&bsh[(p * NC2 + 16 * t + lr) * 2];
            #pragma unroll
            for (int t = 0; t < NT; ++t)
                acc[t] = __builtin_amdgcn_wmma_f32_16x16x4_f32(
                    false, a, false, bb[t], (short)0, acc[t], false, false);
        }
    }

    #pragma unroll
    for (int t = 0; t < NT; ++t) {
        const int col = 16 * t + lr;
        if (col < Nstore) {
            const float bb = bias ? bias[col] : 0.0f;
            #pragma unroll
            for (int j = 0; j < 8; ++j) {          // C: VGPR j -> rows j / j+8
                const int r = bm + j + 8 * half;
                if (r < M) C[(size_t)r * ldC + col] = acc[t][j] + bb;
            }
        }
    }
}

// repack W [K, NCreal] -> chunk-major pair-interleaved [K/32][16][NC2][2]
__global__ void gat_repack_w(const float* __restrict__ Win, float* __restrict__ Wout,
                             int NCreal, int NC2, int total)
{
    const int idx = blockIdx.x * blockDim.x + threadIdx.x;
    if (idx >= total) return;
    const int per   = 16 * NC2 * 2;
    const int chunk = idx / per;
    const int rem   = idx % per;
    const int p     = rem / (NC2 * 2);
    const int rem2  = rem % (NC2 * 2);
    const int c     = rem2 >> 1;
    const int r     = rem2 & 1;
    const int k     = chunk * KC + p * 2 + r;
    Wout[idx] = (c < NCreal) ? Win[k * NCreal + c] : 0.0f;
}

// ---------------------------------------------------------------------------
// Attention logits: als[n,h] = <feat[n,h,:], a_src[h,:]>; H*C == 128.
// ---------------------------------------------------------------------------
__global__ void gat_att_logits(const float* __restrict__ feat,
                               const float* __restrict__ a_src,
                               const float* __restrict__ a_dst,
                               float* __restrict__ als, float* __restrict__ ald,
                               int NH, int H, int C)
{
    const int t = blockIdx.x * blockDim.x + threadIdx.x;
    if (t >= NH) return;
    const int n = t / H, hh = t % H;
    const float* hp = feat + (size_t)n * (H * C) + (size_t)hh * C;
    const float* as = a_src + hh * C;
    const float* ad = a_dst + hh * C;
    float s = 0.f, d = 0.f;
    for (int c = 0; c < C; ++c) { const float v = hp[c]; s += v * as[c]; d += v * ad[c]; }
    als[t] = s; ald[t] = d;
}

__device__ __forceinline__ float lrelu(float x) { return x > 0.f ? x : NEG_SLOPE * x; }

__device__ __forceinline__ void atomicMaxF(float* addr, float val)
{
    if (val >= 0.f) atomicMax((int*)addr, __float_as_int(val));
    else            atomicMin((unsigned int*)addr, __float_as_uint(val));
}

// --- segment-softmax pass 1: init with self-loop edge, then edge atomic max -
__global__ void gat_self_max(const float* __restrict__ als, const float* __restrict__ ald,
                             float* __restrict__ m, int NH)
{
    const int t = blockIdx.x * blockDim.x + threadIdx.x;
    if (t < NH) m[t] = lrelu(als[t] + ald[t]);
}

__global__ void gat_edge_max(const long long* __restrict__ src, const long long* __restrict__ dst,
                             const float* __restrict__ als, const float* __restrict__ ald,
                             float* __restrict__ m, int E, int H)
{
    const int e = blockIdx.x * blockDim.x + threadIdx.x;
    if (e >= E) return;
    const long long s = src[e], d = dst[e];
    for (int h = 0; h < H; ++h)
        atomicMaxF(&m[d * H + h], lrelu(als[s * H + h] + ald[d * H + h]));
}

// --- pass 2: denominators ---------------------------------------------------
__global__ void gat_self_den(const float* __restrict__ als, const float* __restrict__ ald,
                             const float* __restrict__ m, float* __restrict__ den, int NH)
{
    const int t = blockIdx.x * blockDim.x + threadIdx.x;
    if (t < NH) den[t] = __expf(lrelu(als[t] + ald[t]) - m[t]);
}

__global__ void gat_edge_den(const long long* __restrict__ src, const long long* __restrict__ dst,
                             const float* __restrict__ als, const float* __restrict__ ald,
                             const float* __restrict__ m, float* __restrict__ den, int E, int H)
{
    const int e = blockIdx.x * blockDim.x + threadIdx.x;
    if (e >= E) return;
    const long long s = src[e], d = dst[e];
    for (int h = 0; h < H; ++h) {
        const float ex = __expf(lrelu(als[s * H + h] + ald[d * H + h]) - m[d * H + h]);
        atomicAdd(&den[d * H + h], ex);
    }
}

// --- pass 3: init out with self-loop contribution, then edge scatter --------
__global__ void gat_self_agg(const float* __restrict__ feat,
                             const float* __restrict__ als, const float* __restrict__ ald,
                             const float* __restrict__ m, const float* __restrict__ den,
                             float* __restrict__ out, int N, int H, int C)
{
    const int t = blockIdx.x * blockDim.x + threadIdx.x;   // over N*128
    if (t >= N * 128) return;
    const int n = t >> 7, c = t & 127;
    const int hh = c / C;
    const float e = lrelu(als[n * H + hh] + ald[n * H + hh]);
    const float w = __expf(e - m[n * H + hh]) / den[n * H + hh];
    out[t] = w * feat[t];
}

__global__ void gat_edge_agg(const long long* __restrict__ src, const long long* __restrict__ dst,
                             const float* __restrict__ als, const float* __restrict__ ald,
                             const float* __restrict__ m, const float* __restrict__ den,
                             const float* __restrict__ feat, float* __restrict__ out,
                             int E, int H, int C)
{
    const int t = blockIdx.x * blockDim.x + threadIdx.x;   // E * 32 threads
    if (t >= E * 32) return;
    const int e = t >> 5;
    const int c = (t & 31) * 4;                            // 4 consecutive channels
    const long long s = src[e], d = dst[e];
    const int hh = c / C;
    const float lg = lrelu(als[s * H + hh] + ald[d * H + hh]);
    const float alpha = __expf(lg - m[d * H + hh]) / den[d * H + hh];
    const float4 hv = *(const float4*)(feat + (size_t)s * 128 + c);
    float* op = out + (size_t)d * 128 + c;
    atomicAdd(op + 0, alpha * hv.x);
    atomicAdd(op + 1, alpha * hv.y);
    atomicAdd(op + 2, alpha * hv.z);
    atomicAdd(op + 3, alpha * hv.w);
}

// --- elementwise epilogues --------------------------------------------------
__global__ void gat_bias_bn_elu(const float* __restrict__ in, const float* __restrict__ bias,
                                const float* __restrict__ gamma, const float* __restrict__ beta,
                                const float* __restrict__ mean, const float* __restrict__ var,
                                float* __restrict__ out, int total)
{
    const int t = blockIdx.x * blockDim.x + threadIdx.x;
    if (t >= total) return;
    const int c = t & 127;
    float v = in[t] + bias[c];
    v = (v - mean[c]) * __frsqrt_rn(var[c] + BN_EPS) * gamma[c] + beta[c];
    out[t] = v > 0.f ? v : (__expf(v) - 1.f);
}

__global__ void gat_bias_elu(const float* __restrict__ in, const float* __restrict__ bias,
                             float* __restrict__ out, int total)
{
    const int t = blockIdx.x * blockDim.x + threadIdx.x;
    if (t >= total) return;
    float v = in[t] + bias[t & 127];
    out[t] = v > 0.f ? v : (__expf(v) - 1.f);
}

// ---------------------------------------------------------------------------
extern "C" void kernel_launch(void* const* d_in, const int* in_sizes, int n_in,
                              void* d_out, int out_size, void* d_ws, size_t ws_size,
                              hipStream_t stream)
{
    const float*     x    = (const float*)d_in[0];
    const long long* ei   = (const long long*)d_in[1];   // int64 [2, E]
    const float*     W1   = (const float*)d_in[2];
    const float*     as1  = (const float*)d_in[3];
    const float*     ad1  = (const float*)d_in[4];
    const float*     b1   = (const float*)d_in[5];
    const float*     bng  = (const float*)d_in[6];
    const float*     bnb  = (const float*)d_in[7];
    const float*     bnm  = (const float*)d_in[8];
    const float*     bnv  = (const float*)d_in[9];
    const float*     W2   = (const float*)d_in[10];
    const float*     as2  = (const float*)d_in[11];
    const float*     ad2  = (const float*)d_in[12];
    const float*     b2   = (const float*)d_in[13];
    const float*     Wc   = (const float*)d_in[14];
    const float*     bc   = (const float*)d_in[15];
    float*           outp = (float*)d_out;

    const int N = in_sizes[0] / N_CH_IN;                 // 100000 (divisible by 16)
    const int E = in_sizes[1] / 2;                       // 1600000
    const long long* src = ei;
    const long long* dst = ei + E;

    // repacked-weight sizes (floats)
    constexpr int NC2_8  = PadCols<128>::v;              // 144
    constexpr int NC2_3  = PadCols<48>::v;               // 48
    const int W1R = (N_CH_IN / KC) * 16 * NC2_8 * 2;     // 8 chunks
    const int W2R = (HIDC   / KC) * 16 * NC2_8 * 2;      // 4 chunks
    const int WCR = (HIDC   / KC) * 16 * NC2_3 * 2;      // 4 chunks

    // workspace layout (floats)
    float* ws    = (float*)d_ws;
    const size_t NB = (size_t)N * HIDC;
    float* bufH  = ws;                 // h1 -> hbn -> h3
    float* bufA  = bufH + NB;          // agg1 -> h2
    float* bufB  = bufA + NB;          // agg2
    float* als   = bufB + NB;          // [N*4]
    float* ald   = als + (size_t)N * 4;
    float* segm  = ald + (size_t)N * 4;
    float* segd  = segm + (size_t)N * 4;
    float* w1r   = segd + (size_t)N * 4;
    float* w2r   = w1r + W1R;
    float* wcr   = w2r + W2R;

    const int BS = 256;
    auto blk = [](long long n, int b) { return (unsigned)((n + b - 1) / b); };
    const unsigned gRows = blk(N, 128);                  // 128-row stripes

    // ---------------- one-time weight repacks ----------------
    gat_repack_w<<<blk(W1R, BS), BS, 0, stream>>>(W1, w1r, HIDC, NC2_8, W1R);
    gat_repack_w<<<blk(W2R, BS), BS, 0, stream>>>(W2, w2r, HIDC, NC2_8, W2R);
    gat_repack_w<<<blk(WCR, BS), BS, 0, stream>>>(Wc, wcr, OUTC, NC2_3, WCR);

    // ---------------- layer 1 (H=4, C=32) ----------------
    gat_wmma_gemm<8><<<gRows, 256, 0, stream>>>(x, w1r, nullptr, bufH,
                                                N, N_CH_IN, HIDC, HIDC);
    gat_att_logits<<<blk((long long)N * 4, BS), BS, 0, stream>>>(bufH, as1, ad1, als, ald, N * 4, 4, 32);
    gat_self_max<<<blk((long long)N * 4, BS), BS, 0, stream>>>(als, ald, segm, N * 4);
    gat_edge_max<<<blk(E, BS), BS, 0, stream>>>(src, dst, als, ald, segm, E, 4);
    gat_self_den<<<blk((long long)N * 4, BS), BS, 0, stream>>>(als, ald, segm, segd, N * 4);
    gat_edge_den<<<blk(E, BS), BS, 0, stream>>>(src, dst, als, ald, segm, segd, E, 4);
    gat_self_agg<<<blk((long long)N * 128, BS), BS, 0, stream>>>(bufH, als, ald, segm, segd, bufA, N, 4, 32);
    gat_edge_agg<<<blk((long long)E * 32, BS), BS, 0, stream>>>(src, dst, als, ald, segm, segd, bufH, bufA, E, 4, 32);
    gat_bias_bn_elu<<<blk((long long)N * 128, BS), BS, 0, stream>>>(bufA, b1, bng, bnb, bnm, bnv, bufH, N * 128);

    // ---------------- layer 2 (H=1, C=128) ----------------
    gat_wmma_gemm<8><<<gRows, 256, 0, stream>>>(bufH, w2r, nullptr, bufA,
                                                N, HIDC, HIDC, HIDC);
    gat_att_logits<<<blk(N, BS), BS, 0, stream>>>(bufA, as2, ad2, als, ald, N, 1, 128);
    gat_self_max<<<blk(N, BS), BS, 0, stream>>>(als, ald, segm, N);
    gat_edge_max<<<blk(E, BS), BS, 0, stream>>>(src, dst, als, ald, segm, E, 1);
    gat_self_den<<<blk(N, BS), BS, 0, stream>>>(als, ald, segm, segd, N);
    gat_edge_den<<<blk(E, BS), BS, 0, stream>>>(src, dst, als, ald, segm, segd, E, 1);
    gat_self_agg<<<blk((long long)N * 128, BS), BS, 0, stream>>>(bufA, als, ald, segm, segd, bufB, N, 1, 128);
    gat_edge_agg<<<blk((long long)E * 32, BS), BS, 0, stream>>>(src, dst, als, ald, segm, segd, bufA, bufB, E, 1, 128);
    gat_bias_elu<<<blk((long long)N * 128, BS), BS, 0, stream>>>(bufB, b2, bufH, N * 128);

    // ---------------- classifier ----------------
    gat_wmma_gemm<3><<<gRows, 256, 0, stream>>>(bufH, wcr, bc, outp,
                                                N, HIDC, OUTC, OUTC);
}